// HybridEstimatorQNN_65481071403251
// MI455X (gfx1250) — compile-verified
//
#include <hip/hip_runtime.h>

// ---------------------------------------------------------------------------
// HybridEstimatorQNN on MI455X (gfx1250).
//
// The reference output is a SINGLE SCALAR broadcast over BSZ=1M rows:
//   - MLP branch (x,w1,b1,w2,b2) is dead code.
//   - quantum_features() is batch-independent (initial state |0000> for all).
// Optimal plan:
//   Kernel 1 (1 wave): simulate the 4-wire circuit as a chain of 28 complex
//     16x16 matrix products on the matrix pipe (v_wmma_f32_16x16x4_f32),
//     psi = U[:,0]; reduce to the scalar; write to d_ws.
//   Kernel 2: broadcast-fill 4 MiB of f32 output with 128-bit NT stores
//     (~0.18 us at 23.3 TB/s — the true lower bound of this workload).
// ---------------------------------------------------------------------------

typedef __attribute__((ext_vector_type(2))) float v2f;
typedef __attribute__((ext_vector_type(4))) float v4f;
typedef __attribute__((ext_vector_type(8))) float v8f;

// Gate table: {type, wire_or_ctrl, tgt, param_idx}
//   type: 0=rx 1=ry 2=rz 3=cnot
//   param_idx: 0..19 -> rl_params[i], 20 -> rx_theta[0], 21 -> ry_theta[0]
// Entries 0..19 correspond to RL_OPS (np.random.default_rng(0)); table is
// structurally generic and patchable without changing codegen.
static __device__ const int OPS[28][4] = {
    {1, 3, -1, 0},  {3, 2, 0, 1},   {0, 1, -1, 2},  {2, 0, -1, 3},
    {1, 2, -1, 4},  {3, 0, 3, 5},   {0, 3, -1, 6},  {2, 2, -1, 7},
    {1, 0, -1, 8},  {3, 1, 2, 9},   {0, 0, -1, 10}, {1, 1, -1, 11},
    {2, 3, -1, 12}, {3, 3, 1, 13},  {0, 2, -1, 14}, {2, 1, -1, 15},
    {1, 3, -1, 16}, {3, 0, 1, 17},  {0, 0, -1, 18}, {2, 2, -1, 19},
    // final layer: for w in 0..3: rx(rx_theta) then ry(ry_theta)
    {0, 0, -1, 20}, {1, 0, -1, 21}, {0, 1, -1, 20}, {1, 1, -1, 21},
    {0, 2, -1, 20}, {1, 2, -1, 21}, {0, 3, -1, 20}, {1, 3, -1, 21}};

// Element (m,k) of the 16x16 gate matrix. Wire w occupies bit (3-w) of the
// flat index (axis-1 of (2,2,2,2) is the MSB).
__device__ inline void gate_elem(int type, int a, int b, float ch, float sh,
                                 int m, int k, float& gr, float& gi) {
  gr = 0.0f;
  gi = 0.0f;
  if (type == 3) {  // cnot(a=ctrl, b=tgt): out[m] = in[m ^ (mc ? tbit : 0)]
    int mc = (m >> (3 - a)) & 1;
    int j = mc ? (m ^ (1 << (3 - b))) : m;
    gr = (k == j) ? 1.0f : 0.0f;
  } else {
    int msk = (~(1 << (3 - a))) & 15;
    if ((m & msk) != (k & msk)) return;  // other wires must agree
    int mb = (m >> (3 - a)) & 1;
    int kb = (k >> (3 - a)) & 1;
    if (type == 0) {  // rx: [[c, -i s],[-i s, c]]
      if (mb == kb) gr = ch; else gi = -sh;
    } else if (type == 1) {  // ry: [[c, -s],[s, c]]
      if (mb == kb) gr = ch; else gr = mb ? sh : -sh;
    } else {  // rz: diag(e^{-i t/2}, e^{+i t/2})
      if (mb == kb) { gr = ch; gi = mb ? sh : -sh; }
    }
  }
}

// Single-wave circuit simulator. U accumulated as complex 16x16 (Ur,Ui) in
// LDS; each gate applied via 16 chained v_wmma_f32_16x16x4_f32.
__global__ __launch_bounds__(32) void qnn_state_kernel(
    const float* __restrict__ rl_params, const float* __restrict__ rx_theta,
    const float* __restrict__ ry_theta, const float* __restrict__ head_w,
    const float* __restrict__ head_b, float* __restrict__ out_scalar) {
  __shared__ float Ur[16][16];
  __shared__ float Ui[16][16];
  const int lane = threadIdx.x;  // 0..31, wave32
  const int n = lane & 15;       // A-layout M index / B,C,D column index
  const int hi = lane >> 4;      // lane half (selects K sub-pair in A/B)

  // U0 = identity
  for (int i = lane; i < 256; i += 32) {
    Ur[i >> 4][i & 15] = ((i >> 4) == (i & 15)) ? 1.0f : 0.0f;
    Ui[i >> 4][i & 15] = 0.0f;
  }
  __syncthreads();

  for (int g = 0; g < 28; ++g) {
    const int type = OPS[g][0], pa = OPS[g][1], pb = OPS[g][2],
              pi = OPS[g][3];
    const float ang =
        (pi < 20) ? rl_params[pi] : ((pi == 20) ? rx_theta[0] : ry_theta[0]);
    const float ch = cosf(0.5f * ang);
    const float sh = sinf(0.5f * ang);

    // Build A = G (A layout: lane holds row M=n; K chunk c in VGPR pair,
    // lo half K=4c,4c+1, hi half K=4c+2,4c+3) and load B = U from LDS
    // (B layout: lane holds column N=n of rows K as above).
    float ar[8], ai[8], br[8], bi[8];
#pragma unroll
    for (int c = 0; c < 4; ++c) {
      const int k0 = 4 * c + (hi ? 2 : 0);
      const int k1 = k0 + 1;
      gate_elem(type, pa, pb, ch, sh, n, k0, ar[2 * c], ai[2 * c]);
      gate_elem(type, pa, pb, ch, sh, n, k1, ar[2 * c + 1], ai[2 * c + 1]);
      br[2 * c] = Ur[k0][n];
      bi[2 * c] = Ui[k0][n];
      br[2 * c + 1] = Ur[k1][n];
      bi[2 * c + 1] = Ui[k1][n];
    }

    // Complex product D = G * U:
    //   Re(D) = Gr*Ur + (-Gi)*Ui ; Im(D) = Gr*Ui + Gi*Ur
    v8f accR = {0.f, 0.f, 0.f, 0.f, 0.f, 0.f, 0.f, 0.f};
    v8f accI = {0.f, 0.f, 0.f, 0.f, 0.f, 0.f, 0.f, 0.f};
#pragma unroll
    for (int c = 0; c < 4; ++c) {
      v2f Ar = {ar[2 * c], ar[2 * c + 1]};
      v2f Ai = {ai[2 * c], ai[2 * c + 1]};
      v2f nAi = {-ai[2 * c], -ai[2 * c + 1]};
      v2f Br = {br[2 * c], br[2 * c + 1]};
      v2f Bi = {bi[2 * c], bi[2 * c + 1]};
      accR = __builtin_amdgcn_wmma_f32_16x16x4_f32(false, Ar, false, Br,
                                                   (short)0, accR, false, false);
      accR = __builtin_amdgcn_wmma_f32_16x16x4_f32(false, nAi, false, Bi,
                                                   (short)0, accR, false, false);
      accI = __builtin_amdgcn_wmma_f32_16x16x4_f32(false, Ar, false, Bi,
                                                   (short)0, accI, false, false);
      accI = __builtin_amdgcn_wmma_f32_16x16x4_f32(false, Ai, false, Br,
                                                   (short)0, accI, false, false);
    }

    __syncthreads();  // all B reads done before overwrite
    // D layout: VGPR v holds element (M = v + 8*hi, N = n)
#pragma unroll
    for (int v = 0; v < 8; ++v) {
      Ur[v + 8 * hi][n] = accR[v];
      Ui[v + 8 * hi][n] = accI[v];
    }
    __syncthreads();
  }

  if (lane == 0) {
    // psi = U[:,0]; probs -> per-wire <Z>; scalar = feats . head_w + head_b
    float f0 = 0.f, f1 = 0.f, f2 = 0.f, f3 = 0.f;
    for (int m = 0; m < 16; ++m) {
      const float p = Ur[m][0] * Ur[m][0] + Ui[m][0] * Ui[m][0];
      f0 += ((m >> 3) & 1) ? -p : p;  // wire 0 = MSB
      f1 += ((m >> 2) & 1) ? -p : p;
      f2 += ((m >> 1) & 1) ? -p : p;
      f3 += (m & 1) ? -p : p;
    }
    *out_scalar = f0 * head_w[0] + f1 * head_w[1] + f2 * head_w[2] +
                  f3 * head_w[3] + head_b[0];
  }
}

// Bandwidth kernel: broadcast the scalar to 1M f32 with B128 NT stores.
__global__ __launch_bounds__(256) void qnn_bcast_kernel(
    const float* __restrict__ scalar, float* __restrict__ out, int nelem,
    int n4) {
  const int i = blockIdx.x * blockDim.x + threadIdx.x;
  const float v = *scalar;  // uniform -> s_load, L2-resident
  if (i < n4) {
    const v4f v4 = {v, v, v, v};
    __builtin_nontemporal_store(v4, reinterpret_cast<v4f*>(out) + i);
  }
  if (i == 0) {  // tail (none for BSZ=1M, kept for generality)
    for (int t = n4 * 4; t < nelem; ++t) out[t] = v;
  }
}

extern "C" void kernel_launch(void* const* d_in, const int* in_sizes, int n_in,
                              void* d_out, int out_size, void* d_ws,
                              size_t ws_size, hipStream_t stream) {
  // setup_inputs order: 0:x 1:w1 2:b1 3:w2 4:b2 5:rl_params 6:rx_theta
  //                     7:ry_theta 8:head_w 9:head_b
  const float* rl_params = (const float*)d_in[5];
  const float* rx_theta = (const float*)d_in[6];
  const float* ry_theta = (const float*)d_in[7];
  const float* head_w = (const float*)d_in[8];
  const float* head_b = (const float*)d_in[9];
  float* scalar = (float*)d_ws;

  qnn_state_kernel<<<1, 32, 0, stream>>>(rl_params, rx_theta, ry_theta, head_w,
                                         head_b, scalar);

  const int n4 = out_size >> 2;
  const int block = 256;
  const int grid = (n4 + block - 1) / block;
  qnn_bcast_kernel<<<grid, block, 0, stream>>>(scalar, (float*)d_out, out_size,
                                               n4);
}